// BilinearInterpolate_7499012899337
// MI455X (gfx1250) — compile-verified
//
#include <hip/hip_runtime.h>
#include <stdint.h>

// 2x bilinear upsample, fp32 NHWC: (8,256,256,32) -> (8,512,512,32)
// Memory-bound (~320 MiB moved, ~13.7 us at 23.3 TB/s). Strategy:
//  - one block per output row (n, oy)
//  - async-copy input rows r0, r1 (32 KiB each, contiguous) into LDS
//    via gfx1250 GLOBAL_LOAD_ASYNC_TO_LDS_B128 (+ s_wait_asynccnt)
//  - blend from LDS with ds_load_b128, coalesced NON-TEMPORAL b128 stores
//    (output is write-once; NT keeps the 64 MiB input resident in the
//     192 MB L2 so the cross-block row reuse stays an L2 hit)

typedef float v4f __attribute__((ext_vector_type(4)));

#define IN_W    256
#define OUT_WD  512
#define NCH     32
#define ROW_V4  (IN_W * NCH / 4)    // 2048 float4 per input row (32 KiB)
#define OROW_V4 (OUT_WD * NCH / 4)  // 4096 float4 per output row (64 KiB)

__global__ __launch_bounds__(256) void bilinear2x_async_kernel(
    const float* __restrict__ img, float* __restrict__ out)
{
    __shared__ v4f lds[2 * ROW_V4];  // 64 KiB: row r0 then row r1

    const int tid = threadIdx.x;
    const int oy  = blockIdx.x;   // 0..511
    const int n   = blockIdx.y;   // 0..7

    // Vertical coordinate (matches reference: Y = max((oy+0.5)*0.5 - 0.5, 0))
    float Yf = fmaxf(fmaf((float)oy, 0.5f, -0.25f), 0.0f);
    int   r0 = (int)Yf;                         // floor, Yf >= 0
    float h0 = Yf - (float)r0;
    float h1 = 1.0f - h0;
    int   r1 = r0 + ((r0 < IN_W - 1) ? 1 : 0);

    const v4f* g0 = (const v4f*)img + (size_t)(n * IN_W + r0) * ROW_V4;
    const v4f* g1 = (const v4f*)img + (size_t)(n * IN_W + r1) * ROW_V4;

    unsigned lbase0 = (unsigned)(uintptr_t)(&lds[0]);
    unsigned lbase1 = (unsigned)(uintptr_t)(&lds[ROW_V4]);

    // Stage both input rows into LDS with async b128 copies.
    // 2048 float4 per row / 256 threads = 8 async ops per row per thread.
#pragma unroll
    for (int k = 0; k < 8; ++k) {
        int idx = tid + k * 256;
        unsigned l0 = lbase0 + (unsigned)idx * 16u;
        unsigned l1 = lbase1 + (unsigned)idx * 16u;
        uint64_t a0 = (uint64_t)(uintptr_t)(g0 + idx);
        uint64_t a1 = (uint64_t)(uintptr_t)(g1 + idx);
        asm volatile("global_load_async_to_lds_b128 %0, %1, off"
                     :: "v"(l0), "v"(a0) : "memory");
        asm volatile("global_load_async_to_lds_b128 %0, %1, off"
                     :: "v"(l1), "v"(a1) : "memory");
    }
    // Wait for this wave's async fills, then sync the workgroup.
    asm volatile("s_wait_asynccnt 0x0" ::: "memory");
    __syncthreads();

    const v4f* row0 = &lds[0];
    const v4f* row1 = &lds[ROW_V4];
    v4f* orow = (v4f*)out + ((size_t)n * OUT_WD + (size_t)oy) * OROW_V4;

    // 4096 output float4 per row / 256 threads = 16 per thread.
#pragma unroll 4
    for (int k = 0; k < 16; ++k) {
        int j  = tid + k * 256;      // float4 index within output row
        int ox = j >> 3;             // output column (8 float4 per pixel)
        int c4 = j & 7;              // channel-quad index

        float Xf = fmaxf(fmaf((float)ox, 0.5f, -0.25f), 0.0f);
        int   c0 = (int)Xf;
        float w0 = Xf - (float)c0;
        float w1 = 1.0f - w0;
        int   c1 = c0 + ((c0 < IN_W - 1) ? 1 : 0);

        v4f v00 = row0[c0 * 8 + c4];
        v4f v01 = row0[c1 * 8 + c4];
        v4f v10 = row1[c0 * 8 + c4];
        v4f v11 = row1[c1 * 8 + c4];

        // Same association as reference: h1*(w1*v00 + w0*v01) + h0*(w1*v10 + w0*v11)
        v4f res = h1 * (w1 * v00 + w0 * v01) + h0 * (w1 * v10 + w0 * v11);
        __builtin_nontemporal_store(res, &orow[j]);   // global_store_b128 th:NT
    }
}

extern "C" void kernel_launch(void* const* d_in, const int* in_sizes, int n_in,
                              void* d_out, int out_size, void* d_ws, size_t ws_size,
                              hipStream_t stream) {
    const float* img = (const float*)d_in[0];
    float* out = (float*)d_out;
    dim3 grid(OUT_WD, 8);   // (oy, n)
    bilinear2x_async_kernel<<<grid, 256, 0, stream>>>(img, out);
}